// HGCF_3221225472207
// MI455X (gfx1250) — compile-verified
//
#include <hip/hip_runtime.h>
#include <hip/hip_bf16.h>

// ---------------------------------------------------------------------------
// Problem constants (match reference setup_inputs)
// ---------------------------------------------------------------------------
#define NN 50000          // nodes
#define CC 64             // classes
#define GG 3              // metapath graphs
#define EE 1200000        // edges per graph
#define LL 2              // propagation layers
#define DD 256            // input feature dim
#define HH 256            // hidden dim

typedef __attribute__((ext_vector_type(2))) float v2f;
typedef __attribute__((ext_vector_type(8))) float v8f;

// ---------------------------------------------------------------------------
// Small utility kernels
// ---------------------------------------------------------------------------
__global__ void fill_kernel(float* __restrict__ p, float v, long n) {
    long i = (long)blockIdx.x * blockDim.x + threadIdx.x;
    if (i < n) p[i] = v;
}

__global__ void copy4_kernel(float4* __restrict__ dst, const float4* __restrict__ src, long n4) {
    long i = (long)blockIdx.x * blockDim.x + threadIdx.x;
    if (i < n4) dst[i] = src[i];
}

// masked_label = 1 - train_mask ; masked_oh = labels_oh * train_mask ;
// att_sm = softmax(attention, axis=1)
__global__ void node_prep_kernel(const float* __restrict__ train_mask,
                                 const float* __restrict__ labels_oh,
                                 const float* __restrict__ attention,
                                 float* __restrict__ masked_label,
                                 float* __restrict__ masked_oh,
                                 float* __restrict__ att_sm) {
    int i = blockIdx.x * blockDim.x + threadIdx.x;
    if (i >= NN) return;
    float tm = train_mask[i];
    masked_label[i] = 1.0f - tm;
    #pragma unroll 4
    for (int c = 0; c < CC; ++c)
        masked_oh[(size_t)i * CC + c] = labels_oh[(size_t)i * CC + c] * tm;
    float a0 = attention[i * GG + 0];
    float a1 = attention[i * GG + 1];
    float a2 = attention[i * GG + 2];
    float mx = fmaxf(a0, fmaxf(a1, a2));
    float e0 = __expf(a0 - mx), e1 = __expf(a1 - mx), e2 = __expf(a2 - mx);
    float inv = 1.0f / (e0 + e1 + e2);
    att_sm[i * GG + 0] = e0 * inv;
    att_sm[i * GG + 1] = e1 * inv;
    att_sm[i * GG + 2] = e2 * inv;
}

// ---------------------------------------------------------------------------
// Edge softmax: atomic float max via monotone int mapping (safe everywhere)
// ---------------------------------------------------------------------------
__device__ __forceinline__ void atomicMaxF32(float* addr, float v) {
    if (v >= 0.0f)
        atomicMax((int*)addr, __float_as_int(v));
    else
        atomicMin((unsigned int*)addr, (unsigned int)__float_as_int(v));
}

__global__ void edge_max_kernel(const float* __restrict__ e,
                                const int* __restrict__ dst,
                                float* __restrict__ m) {
    int i = blockIdx.x * blockDim.x + threadIdx.x;
    if (i >= EE) return;
    atomicMaxF32(&m[dst[i]], e[i]);
}

__global__ void edge_expsum_kernel(const float* __restrict__ e,
                                   const int* __restrict__ dst,
                                   const float* __restrict__ m,
                                   float* __restrict__ ae,
                                   float* __restrict__ s) {
    int i = blockIdx.x * blockDim.x + threadIdx.x;
    if (i >= EE) return;
    int d = dst[i];
    float a = __expf(e[i] - m[d]);
    ae[i] = a;
    atomicAdd(&s[d], a);
}

// One thread per (edge, class). Each wave handles a single edge (uniform
// src/dst/weight loads -> scalarized), lanes cover consecutive classes so
// gathers and scatter-atomics are fully coalesced; target lives in L2.
__global__ __launch_bounds__(256) void scatter_kernel(const float* __restrict__ h,
                                                      const int* __restrict__ src,
                                                      const int* __restrict__ dst,
                                                      const float* __restrict__ ae,
                                                      const float* __restrict__ s,
                                                      float* __restrict__ hn) {
    long t = (long)blockIdx.x * blockDim.x + threadIdx.x;
    if (t >= (long)EE * CC) return;
    int e = (int)(t >> 6);
    int c = (int)(t & (CC - 1));
    int sv = src[e];
    int dv = dst[e];
    float w = ae[e] / s[dv];
    atomicAdd(&hn[(size_t)dv * CC + c], h[(size_t)sv * CC + c] * w);
}

// h = hn * masked_label[node] + masked_oh
__global__ void clamp_kernel(const float* __restrict__ hn,
                             const float* __restrict__ ml,
                             const float* __restrict__ moh,
                             float* __restrict__ h) {
    long i = (long)blockIdx.x * blockDim.x + threadIdx.x;
    if (i >= (long)NN * CC) return;
    int node = (int)(i >> 6);
    h[i] = hn[i] * ml[node] + moh[i];
}

// lp += att_sm[node,g] * h
__global__ void lp_accum_kernel(const float* __restrict__ h,
                                const float* __restrict__ att_sm,
                                float* __restrict__ lp, int g) {
    long i = (long)blockIdx.x * blockDim.x + threadIdx.x;
    if (i >= (long)NN * CC) return;
    int node = (int)(i >> 6);
    lp[i] += h[i] * att_sm[node * GG + g];
}

// logits = sig(alpha)*lp + (1-sig(alpha))*ns ; also emit logits_lp copy
__global__ void final_kernel(const float* __restrict__ lp,
                             const float* __restrict__ ns,
                             const float* __restrict__ alpha,
                             float* __restrict__ out_logits,
                             float* __restrict__ out_lp) {
    long i = (long)blockIdx.x * blockDim.x + threadIdx.x;
    if (i >= (long)NN * CC) return;
    int node = (int)(i >> 6);
    float sa = 1.0f / (1.0f + __expf(-alpha[node]));
    float l = lp[i];
    out_lp[i] = l;
    out_logits[i] = sa * l + (1.0f - sa) * ns[i];
}

// ---------------------------------------------------------------------------
// f32 WMMA GEMM:  C[M,Nn] = act(A[M,K] * B[K,Nn] + bias)
// Block tile 64x64, 8 waves (4 along M, 2 along N), each wave: 16x32 strip
// as two v_wmma_f32_16x16x4_f32 accumulators. K staged via LDS in 32-slabs.
// ---------------------------------------------------------------------------
#define TM 64
#define TN 64
#define TK 32

template <bool RELU>
__global__ __launch_bounds__(256) void gemm_wmma_kernel(const float* __restrict__ A,
                                                        const float* __restrict__ B,
                                                        const float* __restrict__ bias,
                                                        float* __restrict__ C,
                                                        int M, int Nn, int K) {
    __shared__ float As[TM][TK + 4];   // row stride 36 floats (144B, 16B aligned)
    __shared__ float Bs[TK][TN + 4];   // row stride 68 floats (272B, 16B aligned)

    const int tid  = threadIdx.x;
    const int lane = tid & 31;
    const int wid  = tid >> 5;     // 0..7
    const int wm   = wid >> 1;     // 0..3  (M strip)
    const int wn   = wid & 1;      // 0..1  (N strip of 32)

    const int block_row = blockIdx.x * TM;
    const int block_col = blockIdx.y * TN;

    v8f acc0 = {};
    v8f acc1 = {};

    const int am  = wm * 16 + (lane & 15);       // row within A tile
    const int kb  = (lane >> 4) * 2;             // K-pair select per ISA layout
    const int bn0 = wn * 32 + (lane & 15);       // col within B tile

    for (int k0 = 0; k0 < K; k0 += TK) {
        // Cooperative A tile load: 64x32 f32 = 512 float4, 2 per thread
        #pragma unroll
        for (int i = 0; i < 2; ++i) {
            int idx = tid + i * 256;
            int r  = idx >> 3;           // 0..63
            int c4 = (idx & 7) * 4;      // 0..28
            int gr = block_row + r;
            float4 v = make_float4(0.f, 0.f, 0.f, 0.f);
            if (gr < M) v = *(const float4*)(A + (size_t)gr * K + k0 + c4);
            *(float4*)&As[r][c4] = v;
        }
        // Cooperative B tile load: 32x64 f32 = 512 float4, 2 per thread
        #pragma unroll
        for (int i = 0; i < 2; ++i) {
            int idx = tid + i * 256;
            int r  = idx >> 4;           // 0..31
            int c4 = (idx & 15) * 4;     // 0..60
            float4 v = *(const float4*)(B + (size_t)(k0 + r) * Nn + block_col + c4);
            *(float4*)&Bs[r][c4] = v;
        }
        __syncthreads();

        #pragma unroll
        for (int kk = 0; kk < TK; kk += 4) {
            // A fragment (16x4): lanes 0-15 hold K-pair {0,1}, lanes 16-31 {2,3}
            v2f a;
            a.x = As[am][kk + kb];
            a.y = As[am][kk + kb + 1];
            // B fragments (4x16): same K-pair striping, N across lanes 0-15
            v2f b0, b1v;
            b0.x  = Bs[kk + kb][bn0];
            b0.y  = Bs[kk + kb + 1][bn0];
            b1v.x = Bs[kk + kb][bn0 + 16];
            b1v.y = Bs[kk + kb + 1][bn0 + 16];
            acc0 = __builtin_amdgcn_wmma_f32_16x16x4_f32(false, a, false, b0,
                                                         (short)0, acc0, false, false);
            acc1 = __builtin_amdgcn_wmma_f32_16x16x4_f32(false, a, false, b1v,
                                                         (short)0, acc1, false, false);
        }
        __syncthreads();
    }

    // Store: acc element r -> row = base + r (+8 for lanes 16-31), col = lane&15
    const int col0  = block_col + wn * 32 + (lane & 15);
    const int rbase = block_row + wm * 16 + (lane >> 4) * 8;
    const float bi0 = bias[col0];
    const float bi1 = bias[col0 + 16];
    #pragma unroll
    for (int r = 0; r < 8; ++r) {
        int row = rbase + r;
        if (row < M) {
            float c0 = acc0[r] + bi0;
            float c1 = acc1[r] + bi1;
            if (RELU) { c0 = fmaxf(c0, 0.f); c1 = fmaxf(c1, 0.f); }
            C[(size_t)row * Nn + col0]      = c0;
            C[(size_t)row * Nn + col0 + 16] = c1;
        }
    }
}

// ---------------------------------------------------------------------------
// Host-side orchestration
// ---------------------------------------------------------------------------
static inline long cdiv(long a, long b) { return (a + b - 1) / b; }

extern "C" void kernel_launch(void* const* d_in, const int* in_sizes, int n_in,
                              void* d_out, int out_size, void* d_ws, size_t ws_size,
                              hipStream_t stream) {
    (void)in_sizes; (void)n_in; (void)out_size; (void)ws_size;

    const float* features0 = (const float*)d_in[0];
    const float* label_init = (const float*)d_in[1];
    const float* labels_oh = (const float*)d_in[2];
    const float* train_mask = (const float*)d_in[3];
    const int*   src_all = (const int*)d_in[4];     // [G,E]
    const int*   dst_all = (const int*)d_in[5];     // [G,E]
    const float* e_edges = (const float*)d_in[6];   // [L,G,E]
    const float* attention = (const float*)d_in[7]; // [N,G]
    const float* alpha = (const float*)d_in[8];     // [N,1]
    const float* W1 = (const float*)d_in[9];
    const float* b1 = (const float*)d_in[10];
    const float* W2 = (const float*)d_in[11];
    const float* b2 = (const float*)d_in[12];

    const long NC = (long)NN * CC;

    // Workspace carve-up (floats)
    float* ws = (float*)d_ws;
    float* masked_label = ws;                 ws += NN;
    float* att_sm       = ws;                 ws += (long)NN * GG;
    float* mbuf         = ws;                 ws += NN;
    float* sbuf         = ws;                 ws += NN;
    ws += (16 - ((ws - (float*)d_ws) & 15)) & 15;   // realign to 16 floats
    float* masked_oh    = ws;                 ws += NC;
    float* hbuf         = ws;                 ws += NC;
    float* hnext        = ws;                 ws += NC;
    float* lp           = ws;                 ws += NC;
    float* ae           = ws;                 ws += EE;
    ws += (16 - ((ws - (float*)d_ws) & 15)) & 15;
    float* hm           = ws;                 ws += (long)NN * HH;

    float* out_logits = (float*)d_out;
    float* out_lp     = out_logits + NC;
    float* out_ns     = out_logits + 2 * NC;

    const int B = 256;
    const float NEG_INF = __builtin_huge_valf() * -1.0f;

    // Node prep + zero lp accumulator
    node_prep_kernel<<<cdiv(NN, B), B, 0, stream>>>(train_mask, labels_oh, attention,
                                                    masked_label, masked_oh, att_sm);
    fill_kernel<<<cdiv(NC, B), B, 0, stream>>>(lp, 0.0f, NC);

    // Label propagation over the 3 metapath graphs
    for (int g = 0; g < GG; ++g) {
        const int* sg = src_all + (size_t)g * EE;
        const int* dg = dst_all + (size_t)g * EE;

        copy4_kernel<<<cdiv(NC / 4, B), B, 0, stream>>>((float4*)hbuf,
                                                        (const float4*)label_init, NC / 4);
        for (int l = 0; l < LL; ++l) {
            const float* eg = e_edges + ((size_t)l * GG + g) * EE;

            fill_kernel<<<cdiv(NN, B), B, 0, stream>>>(mbuf, NEG_INF, NN);
            fill_kernel<<<cdiv(NN, B), B, 0, stream>>>(sbuf, 0.0f, NN);
            fill_kernel<<<cdiv(NC, B), B, 0, stream>>>(hnext, 0.0f, NC);

            edge_max_kernel<<<cdiv(EE, B), B, 0, stream>>>(eg, dg, mbuf);
            edge_expsum_kernel<<<cdiv(EE, B), B, 0, stream>>>(eg, dg, mbuf, ae, sbuf);
            scatter_kernel<<<cdiv((long)EE * CC, B), B, 0, stream>>>(hbuf, sg, dg,
                                                                     ae, sbuf, hnext);
            clamp_kernel<<<cdiv(NC, B), B, 0, stream>>>(hnext, masked_label,
                                                        masked_oh, hbuf);
        }
        lp_accum_kernel<<<cdiv(NC, B), B, 0, stream>>>(hbuf, att_sm, lp, g);
    }

    // MLP via f32 WMMA GEMMs
    {
        dim3 grid1((unsigned)cdiv(NN, TM), HH / TN);
        gemm_wmma_kernel<true><<<grid1, B, 0, stream>>>(features0, W1, b1, hm,
                                                        NN, HH, DD);
        dim3 grid2((unsigned)cdiv(NN, TM), CC / TN);
        gemm_wmma_kernel<false><<<grid2, B, 0, stream>>>(hm, W2, b2, out_ns,
                                                         NN, CC, HH);
    }

    // Blend and emit (logits, logits_lp, logits_ns)
    final_kernel<<<cdiv(NC, B), B, 0, stream>>>(lp, out_ns, alpha,
                                                out_logits, out_lp);
}